// ConnectedFilterLayer_18339510354276
// MI455X (gfx1250) — compile-verified
//
#include <hip/hip_runtime.h>
#include <cmath>

// ConnectedFilterLayer for MI455X (gfx1250, wave32).
// Phases:
//  1) init: copy parent -> ptr workspace (NT read), zero dummy node value
//  2) gate: logits via V_WMMA_F32_16X16X4_F32 (weight broadcast in A rows,
//     attrs as B columns, NT streaming reads), sigmoid gate * residue -> vals
//  3) 22x pointer-doubling steps (ping-pong, B128 vector streaming; the 64MB
//     cum/ptr working set stays resident in the 192MB L2 — NT hints on all
//     one-shot streams protect that residency)
//  4) scatter vals along pixel_to_node into the 4096x4096 image (NT in/out)

typedef __attribute__((ext_vector_type(2))) float v2f;
typedef __attribute__((ext_vector_type(8))) float v8f;
typedef __attribute__((ext_vector_type(4))) float v4f;
typedef __attribute__((ext_vector_type(4))) int   v4i;

#define N_DOUBLE 22

__global__ void init_ptr_kernel(const int* __restrict__ parent,
                                int* __restrict__ ptrA,
                                float* __restrict__ cumA, int n1) {
  int i4 = (blockIdx.x * blockDim.x + threadIdx.x) * 4;
  if (i4 + 3 < n1) {
    v4i v = __builtin_nontemporal_load((const v4i*)(parent + i4));
    *(v4i*)(ptrA + i4) = v;  // RT store: re-read by doubling phase
  } else {
    for (int i = i4; i < n1; ++i) {
      ptrA[i] = parent[i];
      if (i == n1 - 1) cumA[i] = 0.0f;  // dummy node value
    }
  }
}

__device__ __forceinline__ float fast_sigmoid(float logit) {
  float s = fminf(fmaxf(100.0f * logit, -12.0f), 12.0f);
  float e = __expf(-s);
  return __builtin_amdgcn_rcpf(1.0f + e);  // v_rcp_f32, ~1ulp
}

// gate kernel: 16 nodes per wave-tile via two chained f32 WMMAs (K=4 + K=2pad)
__global__ void gate_wmma_kernel(const float* __restrict__ attrs,
                                 const float* __restrict__ weight,
                                 const float* __restrict__ bias,
                                 const float* __restrict__ residue,
                                 float* __restrict__ vals, int nNodes) {
  const int lane  = threadIdx.x & 31;
  const int wave  = blockIdx.x * (blockDim.x >> 5) + (threadIdx.x >> 5);
  const int nWaves = gridDim.x * (blockDim.x >> 5);
  const float b0 = bias[0];

  // A matrix: every row m holds the weight vector. 16x4 f32 A layout:
  // lanes 0-15: VGPR0=K0, VGPR1=K1 ; lanes 16-31: VGPR0=K2, VGPR1=K3.
  v2f a1, a2;
  if (lane < 16) {
    a1[0] = weight[0]; a1[1] = weight[1];
    a2[0] = weight[4]; a2[1] = weight[5];
  } else {
    a1[0] = weight[2]; a1[1] = weight[3];
    a2[0] = 0.0f;      a2[1] = 0.0f;
  }

  const int col = lane & 15;
  const int nTiles = nNodes >> 4;  // full 16-node tiles (wave-uniform loop)
  for (int t = wave; t < nTiles; t += nWaves) {
    const int n = (t << 4) + col;
    const float* ap = attrs + (size_t)n * 6;
    // B matrix 4x16 column-major striping: lanes 0-15 hold K0,K1 of column
    // n=lane; lanes 16-31 hold K2,K3 of column n=lane-16.  Attrs are read
    // exactly once -> non-temporal (don't evict the L2-resident cum/ptr).
    v2f bv1, bv2;
    if (lane < 16) {
      bv1[0] = __builtin_nontemporal_load(ap + 0);
      bv1[1] = __builtin_nontemporal_load(ap + 1);
      bv2[0] = __builtin_nontemporal_load(ap + 4);
      bv2[1] = __builtin_nontemporal_load(ap + 5);
    } else {
      bv1[0] = __builtin_nontemporal_load(ap + 2);
      bv1[1] = __builtin_nontemporal_load(ap + 3);
      bv2[0] = 0.0f;
      bv2[1] = 0.0f;
    }
    v8f c = {};
    c = __builtin_amdgcn_wmma_f32_16x16x4_f32(false, a1, false, bv1,
                                              (short)0, c, false, false);
    c = __builtin_amdgcn_wmma_f32_16x16x4_f32(false, a2, false, bv2,
                                              (short)0, c, false, false);
    // D[m][n] identical over m (A rows identical) -> every lane's d[0] is
    // the logit of node (t*16 + lane%16).
    float g = fast_sigmoid(c[0] + b0);
    if (lane < 16) {
      float r = __builtin_nontemporal_load(residue + n);
      vals[n] = g * r;  // RT store: re-read by doubling phase
    }
  }

  // remainder nodes (nNodes % 16), scalar path (empty for N=4,000,000)
  int rem_start = nTiles << 4;
  int gid = blockIdx.x * blockDim.x + threadIdx.x;
  int ri = rem_start + gid;
  if (ri < nNodes) {
    const float* ap = attrs + (size_t)ri * 6;
    float logit = b0;
#pragma unroll
    for (int k = 0; k < 6; ++k) logit += ap[k] * weight[k];
    vals[ri] = fast_sigmoid(logit) * residue[ri];
  }
}

// one pointer-doubling step: cum' = cum + cum[ptr]; ptr' = ptr[ptr]
// All traffic here is L2-resident (2x 32MB ping-pong < 192MB L2): default RT.
__global__ void double_step_kernel(const float* __restrict__ cin,
                                   const int* __restrict__ pin,
                                   float* __restrict__ cout,
                                   int* __restrict__ pout, int n1) {
  int i4 = (blockIdx.x * blockDim.x + threadIdx.x) * 4;
  if (i4 + 3 < n1) {
    v4i p = *(const v4i*)(pin + i4);
    v4f c = *(const v4f*)(cin + i4);
    v4f r; v4i q;
    r[0] = c[0] + cin[p[0]]; q[0] = pin[p[0]];
    r[1] = c[1] + cin[p[1]]; q[1] = pin[p[1]];
    r[2] = c[2] + cin[p[2]]; q[2] = pin[p[2]];
    r[3] = c[3] + cin[p[3]]; q[3] = pin[p[3]];
    *(v4f*)(cout + i4) = r;
    *(v4i*)(pout + i4) = q;
  } else {
    for (int i = i4; i < n1; ++i) {
      int p = pin[i];
      cout[i] = cin[i] + cin[p];
      pout[i] = pin[p];
    }
  }
}

__global__ void scatter_kernel(const float* __restrict__ cum,
                               const int* __restrict__ p2n,
                               float* __restrict__ out, int n) {
  int i4 = (blockIdx.x * blockDim.x + threadIdx.x) * 4;
  if (i4 + 3 < n) {
    v4i p = __builtin_nontemporal_load((const v4i*)(p2n + i4));
    v4f r;
    r[0] = cum[p[0]]; r[1] = cum[p[1]]; r[2] = cum[p[2]]; r[3] = cum[p[3]];
    __builtin_nontemporal_store(r, (v4f*)(out + i4));
  } else {
    for (int i = i4; i < n; ++i) out[i] = cum[p2n[i]];
  }
}

extern "C" void kernel_launch(void* const* d_in, const int* in_sizes, int n_in,
                              void* d_out, int out_size, void* d_ws, size_t ws_size,
                              hipStream_t stream) {
  const float* attrs   = (const float*)d_in[0];
  const float* weight  = (const float*)d_in[1];
  const float* bias    = (const float*)d_in[2];
  const float* residue = (const float*)d_in[3];
  const int*   parent  = (const int*)d_in[4];
  const int*   p2n     = (const int*)d_in[5];
  float*       out     = (float*)d_out;

  const int nNodes = in_sizes[3];  // 4,000,000
  const int n1     = in_sizes[4];  // nNodes + 1
  const int nPix   = in_sizes[5];  // 4096*4096

  // workspace layout: cumA | cumB | ptrA | ptrB  (each n1 padded to 256B)
  size_t n1pad = ((size_t)n1 + 63) & ~(size_t)63;
  char* ws = (char*)d_ws;
  float* cumA = (float*)(ws);
  float* cumB = (float*)(ws + n1pad * 4);
  int*   ptrA = (int*)  (ws + n1pad * 8);
  int*   ptrB = (int*)  (ws + n1pad * 12);

  {
    int threads = 256;
    int blocks = (n1 + threads * 4 - 1) / (threads * 4);
    init_ptr_kernel<<<blocks, threads, 0, stream>>>(parent, ptrA, cumA, n1);
  }
  {
    // 2048 blocks x 8 waves = 16384 waves; each loops over ~15 tiles
    gate_wmma_kernel<<<2048, 256, 0, stream>>>(attrs, weight, bias, residue,
                                               cumA, nNodes);
  }
  {
    float* cin = cumA; float* cout_ = cumB;
    int*   pin = ptrA; int*   pout  = ptrB;
    int threads = 256;
    int blocks = (n1 + threads * 4 - 1) / (threads * 4);
    for (int it = 0; it < N_DOUBLE; ++it) {
      double_step_kernel<<<blocks, threads, 0, stream>>>(cin, pin, cout_, pout, n1);
      float* tc = cin; cin = cout_; cout_ = tc;
      int*   tp = pin; pin = pout;  pout  = tp;
    }
    // N_DOUBLE is even -> final cum is back in cumA (== cin here)
    int t2 = 256;
    int b2 = (nPix + t2 * 4 - 1) / (t2 * 4);
    scatter_kernel<<<b2, t2, 0, stream>>>(cin, p2n, out, nPix);
  }
}